// ChamferLoss_8993661518169
// MI455X (gfx1250) — compile-verified
//
#include <hip/hip_runtime.h>
#include <stdint.h>

typedef float v2f __attribute__((ext_vector_type(2)));
typedef float v8f __attribute__((ext_vector_type(8)));

#define BATCH    4
#define NJC      4          // pred-range chunks
#define CT_MAX   64         // max tiles per chunk (Npred/16/NJC = 64)
#define INF_BITS 0x7F800000

// ---------------------------------------------------------------------------
// Kernel 0: minPred[B*Npred] = minGt[B*Ngt] = +inf (as int bits)
// ---------------------------------------------------------------------------
__global__ void chamfer_init_kernel(int* __restrict__ ws, int n) {
    int idx = blockIdx.x * blockDim.x + threadIdx.x;
    if (idx < n) ws[idx] = INF_BITS;
}

// ---------------------------------------------------------------------------
// Kernel 1: grid = B x (Ngt/128 strip-groups) x NJC pred-chunks.
// Each 256-thread WG: stages its 1024-point pred chunk to LDS as
// float4(x,y,z,ry); each of its 8 waves owns a 16-row gt strip and sweeps the
// 64 pred tiles with v_wmma_f32_16x16x4_f32.
//   P[i][j] = rx[i] + ry[j] - 2*zz[i][j],  rx = +inf for masked-out gt rows.
// Column mins and row mins both accumulate via no-return ds_min_i32 and are
// flushed once per WG with global atomicMin.
// ---------------------------------------------------------------------------
__global__ __launch_bounds__(256)
void chamfer_main_kernel(const float* __restrict__ preds,
                         const float* __restrict__ gts,
                         const uint8_t* __restrict__ mask,
                         int* __restrict__ minPred,   // [B*Npred] bits
                         int* __restrict__ minGt,     // [B*Ngt]   bits
                         int Ngt, int Npred) {
    __shared__ float4 ldsPred[CT_MAX * 16];   // 16 KB: x,y,z,ry per pred point
    __shared__ int    ldsColMin[CT_MAX * 16]; // 4 KB
    __shared__ int    ldsRowMin[128];         // 8 strips * 16 rows

    const int itiles = Ngt >> 4;
    const int jtiles = Npred >> 4;
    const int ct     = jtiles / NJC;          // tiles per chunk (64)
    const int nIG    = itiles >> 3;           // strip groups (32)

    // block decomposition: b, strip-group, chunk
    const int jc  = blockIdx.x % NJC;
    int tmp       = blockIdx.x / NJC;
    const int ig  = tmp % nIG;
    const int b   = tmp / nIG;

    const int lane = threadIdx.x & 31;
    const int wid  = threadIdx.x >> 5;
    const int row  = lane & 15;
    const bool hi  = lane >= 16;

    const int jcbase = jc * ct * 16;          // first pred index of chunk
    const int ibase  = (ig * 8 + wid) * 16;   // first gt row of this wave

    // ---- stage pred chunk: 4 points per thread, ry precomputed into .w
    {
        const float4* src = (const float4*)(preds + ((size_t)b * Npred + jcbase) * 3);
        int t = threadIdx.x;                  // points 4t..4t+3
        float4 q0 = src[t * 3 + 0];
        float4 q1 = src[t * 3 + 1];
        float4 q2 = src[t * 3 + 2];
        float4 p0 = {q0.x, q0.y, q0.z, q0.x*q0.x + q0.y*q0.y + q0.z*q0.z};
        float4 p1 = {q0.w, q1.x, q1.y, q0.w*q0.w + q1.x*q1.x + q1.y*q1.y};
        float4 p2 = {q1.z, q1.w, q2.x, q1.z*q1.z + q1.w*q1.w + q2.x*q2.x};
        float4 p3 = {q2.y, q2.z, q2.w, q2.y*q2.y + q2.z*q2.z + q2.w*q2.w};
        ldsPred[t * 4 + 0] = p0;
        ldsPred[t * 4 + 1] = p1;
        ldsPred[t * 4 + 2] = p2;
        ldsPred[t * 4 + 3] = p3;
        // init min arrays
        for (int i = threadIdx.x; i < ct * 16; i += 256) ldsColMin[i] = INF_BITS;
        if (threadIdx.x < 128) ldsRowMin[threadIdx.x] = INF_BITS;
    }
    __syncthreads();

    // ---- gt strip: A fragment + per-row squared norms (inf for masked rows)
    const float*   gt = gts  + ((size_t)b * Ngt + ibase) * 3;
    const uint8_t* mk = mask + (size_t)b * Ngt + ibase;
    float gx = gt[row * 3 + 0];
    float gy = gt[row * 3 + 1];
    float gz = gt[row * 3 + 2];
    float rx = (mk[row] != 0) ? (gx * gx + gy * gy + gz * gz) : __builtin_inff();

    v2f afrag;
    afrag[0] = hi ? gz : gx;
    afrag[1] = hi ? 0.0f : gy;

    // distribute rx of rows (hi?8:0)+r to every lane
    const int mrow0 = hi ? 8 : 0;
    float rxv[8];
#pragma unroll
    for (int r = 0; r < 8; ++r)
        rxv[r] = __shfl(rx, mrow0 + r, 32);

    float rmin[8];
#pragma unroll
    for (int r = 0; r < 8; ++r) rmin[r] = __builtin_inff();

    // ---- sweep pred tiles of this chunk (LDS-resident, pipelined)
    float4 cur = ldsPred[row];
#pragma unroll 2
    for (int jt = 0; jt < ct; ++jt) {
        int njt = (jt + 1 < ct) ? jt + 1 : jt;
        float4 nxt = ldsPred[njt * 16 + row];

        v2f bfrag;
        bfrag[0] = hi ? cur.z : cur.x;
        bfrag[1] = hi ? 0.0f : cur.y;

        v8f c = {};
        c = __builtin_amdgcn_wmma_f32_16x16x4_f32(
                false, afrag, false, bfrag, (short)0, c, false, false);

        const float ry = cur.w;
        float cmin = __builtin_inff();
#pragma unroll
        for (int r = 0; r < 8; ++r) {
            float p = rxv[r] + ry - 2.0f * c[r];
            rmin[r] = fminf(rmin[r], p);
            cmin    = fminf(cmin, p);
        }
        cmin = fmaxf(cmin, 0.0f);   // >=0 so int-bit ordering == float ordering
        atomicMin(&ldsColMin[jt * 16 + row], __float_as_int(cmin));
        cur = nxt;
    }

    // ---- fold per-lane row mins into LDS (16-way conflicts, no shuffles)
#pragma unroll
    for (int r = 0; r < 8; ++r) {
        float v = fmaxf(rmin[r], 0.0f);      // inf stays inf for masked rows
        atomicMin(&ldsRowMin[wid * 16 + mrow0 + r], __float_as_int(v));
    }

    __syncthreads();

    // ---- flush WG-local mins to global
    int* gcol = minPred + (size_t)b * Npred + jcbase;
    for (int i = threadIdx.x; i < ct * 16; i += 256) {
        int v = ldsColMin[i];
        if (v != INF_BITS) atomicMin(&gcol[i], v);
    }
    if (threadIdx.x < 128) {
        int v = ldsRowMin[threadIdx.x];
        if (v != INF_BITS)
            atomicMin(&minGt[(size_t)b * Ngt + ig * 128 + threadIdx.x], v);
    }
}

// ---------------------------------------------------------------------------
// Kernel 2: out = sum(minPred) + sum(mask ? minGt : 0)
// ---------------------------------------------------------------------------
__global__ __launch_bounds__(256)
void chamfer_reduce_kernel(const int* __restrict__ minPred, int nPred,
                           const int* __restrict__ minGt, int nGt,
                           const uint8_t* __restrict__ mask,
                           float* __restrict__ out) {
    __shared__ float sdata[256];
    float acc = 0.0f;
    for (int i = threadIdx.x; i < nPred; i += 256)
        acc += __int_as_float(minPred[i]);
    for (int i = threadIdx.x; i < nGt; i += 256)
        acc += (mask[i] != 0) ? __int_as_float(minGt[i]) : 0.0f;
    sdata[threadIdx.x] = acc;
    __syncthreads();
    for (int off = 128; off > 0; off >>= 1) {
        if ((int)threadIdx.x < off) sdata[threadIdx.x] += sdata[threadIdx.x + off];
        __syncthreads();
    }
    if (threadIdx.x == 0) out[0] = sdata[0];
}

// ---------------------------------------------------------------------------
extern "C" void kernel_launch(void* const* d_in, const int* in_sizes, int n_in,
                              void* d_out, int out_size, void* d_ws, size_t ws_size,
                              hipStream_t stream) {
    const float*   preds = (const float*)d_in[0];    // [B, Npred, 3]
    const float*   gts   = (const float*)d_in[1];    // [B, Ngt, 3]
    const uint8_t* mask  = (const uint8_t*)d_in[2];  // [B, Ngt] bool

    const int Npred = in_sizes[0] / (BATCH * 3);
    const int Ngt   = in_sizes[1] / (BATCH * 3);

    int* minPred = (int*)d_ws;                       // B*Npred ints
    int* minGt   = minPred + (size_t)BATCH * Npred;  // B*Ngt ints
    const int nPred = BATCH * Npred;
    const int nGt   = BATCH * Ngt;

    // init both min arrays to +inf
    {
        int total  = nPred + nGt;
        int blocks = (total + 255) / 256;
        chamfer_init_kernel<<<blocks, 256, 0, stream>>>(minPred, total);
    }
    // main WMMA sweep: 512 WGs = B * (Ngt/128) * NJC, 8 waves each
    {
        int blocks = BATCH * (Ngt >> 7) * NJC;
        chamfer_main_kernel<<<blocks, 256, 0, stream>>>(
            preds, gts, mask, minPred, minGt, Ngt, Npred);
    }
    // final scalar reduction
    chamfer_reduce_kernel<<<1, 256, 0, stream>>>(
        minPred, nPred, minGt, nGt, mask, (float*)d_out);
}